// RenderModel_24223615549824
// MI455X (gfx1250) — compile-verified
//
#include <hip/hip_runtime.h>
#include <math.h>

// ---------------------------------------------------------------------------
// Softmax forward-splat view synthesis for MI455X (gfx1250, wave32).
//  - per-batch transforms fused to P = Kdst*R*Kinv, tp = Kdst*t
//    (P computed with V_WMMA_F32_16X16X4_F32, 4 batches per WMMA on the
//     block diagonal of a 16x16 tile)
//  - scatter accumulates with hardware GLOBAL_ATOMIC_ADD_F32 directly into
//    the channel-first output planes; metric sum lives in the mask plane
//  - per-batch min/max of 1/z via LDS reduce + ordered-uint atomic min/max
// ---------------------------------------------------------------------------

typedef float v2f __attribute__((ext_vector_type(2)));
typedef float v8f __attribute__((ext_vector_type(8)));

__device__ __forceinline__ void atomAddF(float* p, float v) {
  __hip_atomic_fetch_add(p, v, __ATOMIC_RELAXED, __HIP_MEMORY_SCOPE_AGENT);
}
__device__ __forceinline__ unsigned f2ord(float f) {
  unsigned u = __float_as_uint(f);
  return (u & 0x80000000u) ? ~u : (u | 0x80000000u);
}
__device__ __forceinline__ float ord2f(unsigned o) {
  unsigned u = (o & 0x80000000u) ? (o & 0x7FFFFFFFu) : ~o;
  return __uint_as_float(u);
}

// ---------------------------------------------------------------------------
// setup: per-batch  T1 = R * inv(Ksrc)  (scalar, 8 threads),
// then  P = Kdst * T1  via two V_WMMA_F32_16X16X4_F32 (4 batches each on the
// block diagonal).  One wave; WMMA executed by all 32 lanes (EXEC all ones).
// params[b*16 + 0..8] = P row-major, [9..11] = tp, [12] = scale, [13] = bias.
// ---------------------------------------------------------------------------
__global__ __launch_bounds__(32) void setup_kernel(
    const float* __restrict__ r_cam, const float* __restrict__ t_cam,
    const float* __restrict__ k_src, const float* __restrict__ k_dst,
    float* __restrict__ params, unsigned* __restrict__ mm, int B) {
  __shared__ float T1[8][9];
  __shared__ float Ptile[2][16][16];
  const int lane = threadIdx.x;

  float tp0 = 0.f, tp1 = 0.f, tp2 = 0.f;
  if (lane < B && lane < 8) {
    const float* K = k_src + lane * 9;
    float a = K[0], bb = K[1], c = K[2], d = K[3], e = K[4], f = K[5];
    float g = K[6], h = K[7], i = K[8];
    float A_ = e * i - f * h, B_ = -(d * i - f * g), C_ = d * h - e * g;
    float rd = 1.0f / (a * A_ + bb * B_ + c * C_);
    float inv[9];
    inv[0] = A_ * rd;              inv[1] = -(bb * i - c * h) * rd;
    inv[2] = (bb * f - c * e) * rd;
    inv[3] = B_ * rd;              inv[4] = (a * i - c * g) * rd;
    inv[5] = -(a * f - c * d) * rd;
    inv[6] = C_ * rd;              inv[7] = -(a * h - bb * g) * rd;
    inv[8] = (a * e - bb * d) * rd;
    const float* R = r_cam + lane * 9;
    for (int ii = 0; ii < 3; ++ii)
      for (int jj = 0; jj < 3; ++jj) {
        float s = 0.f;
        for (int kk = 0; kk < 3; ++kk) s = fmaf(R[ii * 3 + kk], inv[kk * 3 + jj], s);
        T1[lane][ii * 3 + jj] = s;
      }
    const float* KD = k_dst + lane * 9;
    const float* t = t_cam + lane * 3;
    tp0 = KD[0] * t[0] + KD[1] * t[1] + KD[2] * t[2];
    tp1 = KD[3] * t[0] + KD[4] * t[1] + KD[5] * t[2];
    tp2 = KD[6] * t[0] + KD[7] * t[1] + KD[8] * t[2];
  }
  __syncthreads();

  const int rc = lane & 15;        // row index for A, col index for B
  const bool hi = lane >= 16;      // lanes 16-31 carry K=2,3 halves
  for (int g2 = 0; g2 < 2; ++g2) {
    float ax = 0.f, ay = 0.f, bx = 0.f, by = 0.f;
    int bb = 4 * g2 + rc / 3;
    int ij = rc - 3 * (rc / 3);
    if (rc < 12 && bb < B) {
      const float* KD = k_dst + bb * 9 + ij * 3;  // A row = Kdst[bb] row ij
      if (!hi) { ax = KD[0]; ay = KD[1]; }        // K=0,1
      else     { ax = KD[2]; ay = 0.f; }          // K=2 (K=3 zero-padded)
      if (!hi) { bx = T1[bb][0 * 3 + ij]; by = T1[bb][1 * 3 + ij]; }
      else     { bx = T1[bb][2 * 3 + ij]; by = 0.f; }
    }
    v2f Af = {ax, ay};
    v2f Bf = {bx, by};
    v8f Cf = {};
    Cf = __builtin_amdgcn_wmma_f32_16x16x4_f32(
        /*neg_a=*/false, Af, /*neg_b=*/false, Bf,
        /*c_mod=*/(short)0, Cf, /*reuse_a=*/false, /*reuse_b=*/false);
    const int baseM = hi ? 8 : 0;
    const int n = lane & 15;       // C/D: VGPR r -> rows r (lanes 0-15) / r+8
#pragma unroll
    for (int rr = 0; rr < 8; ++rr) Ptile[g2][baseM + rr][n] = Cf[rr];
  }
  __syncthreads();

  if (lane < B && lane < 8) {
    const int g2 = lane >> 2, lb = lane & 3, base = 3 * lb;
    float* pp = params + lane * 16;
    for (int ii = 0; ii < 3; ++ii)
      for (int jj = 0; jj < 3; ++jj)
        pp[ii * 3 + jj] = Ptile[g2][base + ii][base + jj];
    pp[9] = tp0; pp[10] = tp1; pp[11] = tp2;
    mm[2 * lane]     = 0xFFFFFFFFu;  // ordered +inf  (min slot)
    mm[2 * lane + 1] = 0x00000000u;  // ordered -inf  (max slot)
  }
}

// ---------------------------------------------------------------------------
__global__ __launch_bounds__(256) void zero_kernel(float4* __restrict__ p, long n4) {
  long stride = (long)gridDim.x * 256;
  for (long i = (long)blockIdx.x * 256 + threadIdx.x; i < n4; i += stride)
    p[i] = make_float4(0.f, 0.f, 0.f, 0.f);
}

// ---------------------------------------------------------------------------
__global__ __launch_bounds__(256) void minmax_kernel(
    const float* __restrict__ depths, const float* __restrict__ params,
    unsigned* __restrict__ mm, int H, int W, int chunks) {
  const int HW = H * W;
  const int b = blockIdx.x / chunks;
  const int chunk = blockIdx.x - b * chunks;
  const int per = HW / chunks;
  const int base = chunk * per;
  __shared__ float sp[4];
  if (threadIdx.x < 4) {
    const float* pp = params + b * 16;
    sp[threadIdx.x] = (threadIdx.x < 3) ? pp[6 + threadIdx.x] : pp[11];
  }
  __syncthreads();
  const float P20 = sp[0], P21 = sp[1], P22 = sp[2], tpz = sp[3];
  float vmin = __builtin_inff(), vmax = -__builtin_inff();
  const float* dptr = depths + (size_t)b * HW;
  for (int i = threadIdx.x; i < per; i += 256) {
    int p = base + i;
    int y = p / W, x = p - y * W;
    float z = fmaf(dptr[p], fmaf(P20, (float)x, fmaf(P21, (float)y, P22)), tpz);
    float imp = 1.0f / z;
    vmin = fminf(vmin, imp);
    vmax = fmaxf(vmax, imp);
  }
  __shared__ float smin[256], smax[256];
  smin[threadIdx.x] = vmin; smax[threadIdx.x] = vmax;
  __syncthreads();
  for (int s = 128; s > 0; s >>= 1) {
    if (threadIdx.x < s) {
      smin[threadIdx.x] = fminf(smin[threadIdx.x], smin[threadIdx.x + s]);
      smax[threadIdx.x] = fmaxf(smax[threadIdx.x], smax[threadIdx.x + s]);
    }
    __syncthreads();
  }
  if (threadIdx.x == 0) {
    __hip_atomic_fetch_min(&mm[2 * b], f2ord(smin[0]),
                           __ATOMIC_RELAXED, __HIP_MEMORY_SCOPE_AGENT);
    __hip_atomic_fetch_max(&mm[2 * b + 1], f2ord(smax[0]),
                           __ATOMIC_RELAXED, __HIP_MEMORY_SCOPE_AGENT);
  }
}

// ---------------------------------------------------------------------------
__global__ __launch_bounds__(32) void finalize_kernel(
    float* __restrict__ params, const unsigned* __restrict__ mm, int B) {
  int b = threadIdx.x;
  if (b < B) {
    float imin = ord2f(mm[2 * b]);
    float imax = ord2f(mm[2 * b + 1]);
    float scale = 20.0f / (imax - imin + 1e-6f);
    params[b * 16 + 12] = scale;
    params[b * 16 + 13] = fmaf(-imin, scale, -10.0f);  // bias
  }
}

// ---------------------------------------------------------------------------
// splat: one thread per source pixel; 24 GLOBAL_ATOMIC_ADD_F32 scatter into
// the channel-first output planes (feature c0..c3, disp=imp*e, metric=e).
// Requires HW % 256 == 0 so each block is within one batch.
// ---------------------------------------------------------------------------
__global__ __launch_bounds__(256) void splat_kernel(
    const float4* __restrict__ imgs, const float* __restrict__ depths,
    const float* __restrict__ params, float* __restrict__ out,
    int B, int H, int W) {
  const int HW = H * W;
  const int blocksPerBatch = HW >> 8;
  const int b = blockIdx.x / blocksPerBatch;
  __shared__ float sp[14];
  if (threadIdx.x < 14) sp[threadIdx.x] = params[b * 16 + threadIdx.x];
  __syncthreads();

  const int p = (blockIdx.x - b * blocksPerBatch) * 256 + threadIdx.x;
  const int y = p / W, x = p - y * W;
  const size_t g = (size_t)b * HW + p;
  const float4 img = imgs[g];
  const float d = depths[g];
  const float xf = (float)x, yf = (float)y;

  const float px = fmaf(d, fmaf(sp[0], xf, fmaf(sp[1], yf, sp[2])), sp[9]);
  const float py = fmaf(d, fmaf(sp[3], xf, fmaf(sp[4], yf, sp[5])), sp[10]);
  const float pz = fmaf(d, fmaf(sp[6], xf, fmaf(sp[7], yf, sp[8])), sp[11]);

  const float rz = 1.0f / fmaxf(pz, 1e-8f);
  const float tx = px * rz, ty = py * rz;   // target coords (coord + flow)
  const float imp = 1.0f / pz;              // 1/new_z (unclamped, as reference)
  const float e = __expf(fmaf(imp, sp[12], sp[13]));

  const float c0 = img.x * e, c1 = img.y * e, c2 = img.z * e, c3 = img.w * e;
  const float c4 = imp * e, c5 = e;

  const float x0f = floorf(tx), y0f = floorf(ty);
  const float fx1 = tx - x0f, fy1 = ty - y0f;
  const float fx0 = 1.f - fx1, fy0 = 1.f - fy1;
  const int x0 = (int)x0f, y0 = (int)y0f;
  const int x1 = x0 + 1, y1 = y0 + 1;

  float* __restrict__ fplane = out + (size_t)b * 4 * HW;
  float* __restrict__ dplane = out + (size_t)(4 * B + b) * HW;
  float* __restrict__ mplane = out + (size_t)(5 * B + b) * HW;

  auto corner = [&](int xi, int yi, float wc) {
    if ((unsigned)xi < (unsigned)W && (unsigned)yi < (unsigned)H) {
      const size_t off = (size_t)yi * W + xi;
      atomAddF(fplane + off,                  c0 * wc);
      atomAddF(fplane + (size_t)HW + off,     c1 * wc);
      atomAddF(fplane + (size_t)2 * HW + off, c2 * wc);
      atomAddF(fplane + (size_t)3 * HW + off, c3 * wc);
      atomAddF(dplane + off, c4 * wc);
      atomAddF(mplane + off, c5 * wc);
    }
  };
  corner(x0, y0, fx0 * fy0);
  corner(x1, y0, fx1 * fy0);
  corner(x0, y1, fx0 * fy1);
  corner(x1, y1, fx1 * fy1);
}

// ---------------------------------------------------------------------------
__global__ __launch_bounds__(256) void norm_kernel(float* __restrict__ out,
                                                   int B, int H, int W) {
  const int HW = H * W;
  const int blocksPerBatch = HW >> 8;
  const int b = blockIdx.x / blocksPerBatch;
  const int off = (blockIdx.x - b * blocksPerBatch) * 256 + threadIdx.x;
  float* __restrict__ fplane = out + (size_t)b * 4 * HW + off;
  float* __restrict__ dptr = out + (size_t)(4 * B + b) * HW + off;
  float* __restrict__ mptr = out + (size_t)(5 * B + b) * HW + off;
  const float s = *mptr;
  const float r = 1.0f / (s + 1e-7f);
  fplane[0]             *= r;
  fplane[(size_t)HW]     *= r;
  fplane[(size_t)2 * HW] *= r;
  fplane[(size_t)3 * HW] *= r;
  *dptr *= r;
  *mptr = s * r;
}

// ---------------------------------------------------------------------------
extern "C" void kernel_launch(void* const* d_in, const int* in_sizes, int n_in,
                              void* d_out, int out_size, void* d_ws, size_t ws_size,
                              hipStream_t stream) {
  const float* imgs   = (const float*)d_in[0];
  const float* depths = (const float*)d_in[1];
  const float* r_cam  = (const float*)d_in[2];
  const float* t_cam  = (const float*)d_in[3];
  const float* k_src  = (const float*)d_in[4];
  const float* k_dst  = (const float*)d_in[5];

  const int B = in_sizes[3] / 3;                 // t_cam is [B,3]
  const long HW = (long)in_sizes[1] / B;         // depths is [B,H,W]
  const int W = (int)(sqrt((double)HW) + 0.5);   // square frames (1024x1024)
  const int H = (int)(HW / W);

  float* params = (float*)d_ws;                  // B*16 floats
  unsigned* mm = (unsigned*)d_ws + B * 16;       // 2*B ordered uints

  setup_kernel<<<1, 32, 0, stream>>>(r_cam, t_cam, k_src, k_dst, params, mm, B);
  zero_kernel<<<2048, 256, 0, stream>>>((float4*)d_out, (long)out_size / 4);
  const int chunks = 64;
  minmax_kernel<<<B * chunks, 256, 0, stream>>>(depths, params, mm, H, W, chunks);
  finalize_kernel<<<1, 32, 0, stream>>>(params, mm, B);
  const int nblk = (int)((long)B * HW / 256);
  splat_kernel<<<nblk, 256, 0, stream>>>((const float4*)imgs, depths, params,
                                         (float*)d_out, B, H, W);
  norm_kernel<<<nblk, 256, 0, stream>>>((float*)d_out, B, H, W);
}